// GatedStructureBias_55851754717698
// MI455X (gfx1250) — compile-verified
//
#include <hip/hip_runtime.h>
#include <math.h>

typedef float v2f __attribute__((ext_vector_type(2)));
typedef float v4f __attribute__((ext_vector_type(4)));
typedef float v8f __attribute__((ext_vector_type(8)));

#define L_SEQ     512
#define D_STATE   256
#define D_PAIR    128
#define H_DIM     32
#define D_RBF     36

// ---------------------------------------------------------------------------
// Kernel 1: LayerNorm(state) -> left/right projections.  One block per row l.
// ---------------------------------------------------------------------------
__global__ __launch_bounds__(256)
void k_ln_proj(const float* __restrict__ state,
               const float* __restrict__ ln_w, const float* __restrict__ ln_b,
               const float* __restrict__ w_left,  const float* __restrict__ b_left,
               const float* __restrict__ w_right, const float* __restrict__ b_right,
               float* __restrict__ left, float* __restrict__ right) {
  __shared__ float sx[D_STATE];
  __shared__ float sa[D_STATE];
  __shared__ float sb[D_STATE];
  __shared__ float sn[D_STATE];
  const int l = blockIdx.x;
  const int tid = threadIdx.x;

  float x = state[l * D_STATE + tid];
  sx[tid] = x;
  sa[tid] = x;
  sb[tid] = x * x;
  __syncthreads();
  for (int s = D_STATE / 2; s > 0; s >>= 1) {
    if (tid < s) { sa[tid] += sa[tid + s]; sb[tid] += sb[tid + s]; }
    __syncthreads();
  }
  const float mu  = sa[0] * (1.0f / D_STATE);
  const float var = sb[0] * (1.0f / D_STATE) - mu * mu;
  const float inv = rsqrtf(var + 1e-5f);
  sn[tid] = (sx[tid] - mu) * inv * ln_w[tid] + ln_b[tid];
  __syncthreads();

  if (tid < 2 * H_DIM) {
    const int h = tid & (H_DIM - 1);
    const float* __restrict__ W  = (tid < H_DIM) ? w_left  : w_right;
    const float* __restrict__ Bv = (tid < H_DIM) ? b_left  : b_right;
    float acc = Bv[h];
    #pragma unroll 8
    for (int i = 0; i < D_STATE; ++i) acc += sn[i] * W[i * H_DIM + h];
    float* dst = (tid < H_DIM) ? left : right;
    dst[l * H_DIM + h] = acc;
  }
}

// ---------------------------------------------------------------------------
// Kernel 2: t[l,j,p] = sum_i left[l,i] * Wg[i,j,p]   (Wg = w_gate.reshape(32,32,128))
// Output layout is permuted-transposed for WMMA B-frag loads in k_main:
//   t2[ (l*128 + p)*32 + jperm(j) ]
//   jperm groups j%4 in {0,1} into slots [0,16) and j%4 in {2,3} into [16,32),
//   so each WMMA lane reads 16 consecutive floats (4 x b128).
// grid (16, 512): blockIdx.y = l, 256 threads cover 256 of 4096 (j,p) pairs.
// ---------------------------------------------------------------------------
__global__ __launch_bounds__(256)
void k_t(const float* __restrict__ left, const float* __restrict__ w_gate,
         float* __restrict__ t2) {
  __shared__ float sl[H_DIM];
  const int l  = blockIdx.y;
  const int jp = blockIdx.x * 256 + threadIdx.x;     // 0..4095
  if (threadIdx.x < H_DIM) sl[threadIdx.x] = left[l * H_DIM + threadIdx.x];
  __syncthreads();
  const int j = jp >> 7;           // /128
  const int p = jp & (D_PAIR - 1);
  float acc = 0.0f;
  #pragma unroll
  for (int i = 0; i < H_DIM; ++i)
    acc += sl[i] * w_gate[(i * H_DIM + j) * D_PAIR + p];
  const int a = j >> 2, r = j & 3;
  const int jperm = (r < 2) ? (2 * a + r) : (16 + 2 * a + (r - 2));
  t2[((size_t)l * D_PAIR + p) * H_DIM + jperm] = acc;
}

// ---------------------------------------------------------------------------
// Kernel 3 (main, fused): per (16l x 16m) tile:
//   phi (RBF of pairwise distances) -> LDS
//   rbf  = phi @ w_rbf + b_rbf      (WMMA f32 16x16x4, K=36 -> 9 steps)
//   logit= right @ t^T + b_gate     (WMMA f32 16x16x4, K=32 -> 8 steps)
//   out  = sigmoid(logit) * rbf
// 8 waves per block, wave w owns p-chunk [16w, 16w+16).
// ---------------------------------------------------------------------------
__global__ __launch_bounds__(256)
void k_main(const float* __restrict__ xyz, const float* __restrict__ t2,
            const float* __restrict__ right, const float* __restrict__ w_rbf,
            const float* __restrict__ b_rbf, const float* __restrict__ b_gate,
            float* __restrict__ out) {
  __shared__ float phi[16 * 16 * D_RBF];   // 36 KB

  const int tid = threadIdx.x;
  const int l0 = blockIdx.x * 16;
  const int m0 = blockIdx.y * 16;

  // ---- Stage A: distances + RBF features into LDS (one (l,m) pair/thread) ----
  {
    const int lA = tid >> 4, mA = tid & 15;
    const float* ca = xyz + (size_t)(l0 + lA) * 9 + 3;  // atom index 1 of 3
    const float* cb = xyz + (size_t)(m0 + mA) * 9 + 3;
    const float dx = ca[0] - cb[0];
    const float dy = ca[1] - cb[1];
    const float dz = ca[2] - cb[2];
    const float d2 = dx * dx + dy * dy + dz * dz;
    const float dist = sqrtf(fmaxf(d2, 1e-12f));
    const float inv_sigma = 36.0f / 20.0f;              // 1 / ((22-2)/36)
    float* dst = &phi[tid * D_RBF];
    #pragma unroll
    for (int k = 0; k < D_RBF; ++k) {
      const float muk = 2.0f + (float)k * (20.0f / 35.0f);  // linspace(2,22,36)
      const float u = (dist - muk) * inv_sigma;
      dst[k] = __expf(-u * u);
    }
  }
  __syncthreads();

  const int wave = tid >> 5;              // 0..7 -> p chunk
  const int lane = tid & 31;
  const int half = lane >> 4;             // 0: K+0/K+1 rows, 1: K+2/K+3 rows
  const int ml   = lane & 15;             // M (or N) index within tile
  const int p    = wave * 16 + ml;        // this lane's p channel

  // ---- Preload A = right tile (constant for whole block) -------------------
  v2f afr[8];
  #pragma unroll
  for (int kk = 0; kk < 8; ++kk) {
    const int j = 4 * kk + 2 * half;
    afr[kk] = *(const v2f*)(right + (size_t)(m0 + ml) * H_DIM + j);
  }

  // ---- Preload B = w_rbf frags for this wave's p-chunk ---------------------
  v2f bw[9];
  #pragma unroll
  for (int kk = 0; kk < 9; ++kk) {
    const int k0 = 4 * kk + 2 * half;
    bw[kk].x = w_rbf[(size_t)k0 * D_PAIR + p];
    bw[kk].y = w_rbf[(size_t)(k0 + 1) * D_PAIR + p];
  }

  const float bg = b_gate[p];
  const float br = b_rbf[p];

  // ---- Loop over the 16 l-rows of the tile --------------------------------
  for (int l = 0; l < 16; ++l) {
    // Load this lane's 16 t-values (permuted layout): 4 x global_load_b128
    const float* tb = t2 + ((size_t)(l0 + l) * D_PAIR + p) * H_DIM + 16 * half;
    float tf[16];
    *(v4f*)&tf[0]  = *(const v4f*)(tb + 0);
    *(v4f*)&tf[4]  = *(const v4f*)(tb + 4);
    *(v4f*)&tf[8]  = *(const v4f*)(tb + 8);
    *(v4f*)&tf[12] = *(const v4f*)(tb + 12);

    // rbf[m,p] = sum_k phi[l,m,k] * w_rbf[k,p]
    v8f racc = {0.f, 0.f, 0.f, 0.f, 0.f, 0.f, 0.f, 0.f};
    #pragma unroll
    for (int kk = 0; kk < 9; ++kk) {
      const v2f ap = *(const v2f*)&phi[(l * 16 + ml) * D_RBF + 4 * kk + 2 * half];
      racc = __builtin_amdgcn_wmma_f32_16x16x4_f32(
          false, ap, false, bw[kk], (short)0, racc, false, false);
    }

    // logit[m,p] = sum_j right[m,j] * t[l,j,p]
    v8f lacc = {0.f, 0.f, 0.f, 0.f, 0.f, 0.f, 0.f, 0.f};
    #pragma unroll
    for (int kk = 0; kk < 8; ++kk) {
      v2f bt;
      bt.x = tf[2 * kk];
      bt.y = tf[2 * kk + 1];
      lacc = __builtin_amdgcn_wmma_f32_16x16x4_f32(
          false, afr[kk], false, bt, (short)0, lacc, false, false);
    }

    // Epilogue: sigmoid gate * rbf, coalesced stores (16 consecutive p/half-wave)
    // sigmoid = rcp(1 + exp(-x)) via hardware v_rcp_f32 (avoids IEEE div sequence)
    float* obase = out + ((size_t)(l0 + l) * L_SEQ + (m0 + half * 8)) * D_PAIR + p;
    #pragma unroll
    for (int g = 0; g < 8; ++g) {
      const float logit = lacc[g] + bg;
      const float e     = __expf(-logit);
      const float gate  = __builtin_amdgcn_rcpf(1.0f + e);
      obase[(size_t)g * D_PAIR] = gate * (racc[g] + br);
    }
  }
}

// ---------------------------------------------------------------------------
extern "C" void kernel_launch(void* const* d_in, const int* in_sizes, int n_in,
                              void* d_out, int out_size, void* d_ws, size_t ws_size,
                              hipStream_t stream) {
  const float* xyz     = (const float*)d_in[0];
  const float* state   = (const float*)d_in[1];
  const float* ln_w    = (const float*)d_in[2];
  const float* ln_b    = (const float*)d_in[3];
  const float* w_rbf   = (const float*)d_in[4];
  const float* b_rbf   = (const float*)d_in[5];
  const float* w_left  = (const float*)d_in[6];
  const float* b_left  = (const float*)d_in[7];
  const float* w_right = (const float*)d_in[8];
  const float* b_right = (const float*)d_in[9];
  const float* w_gate  = (const float*)d_in[10];
  const float* b_gate  = (const float*)d_in[11];
  float* out = (float*)d_out;

  float* ws    = (float*)d_ws;
  float* left  = ws;                                  // 512*32
  float* right = ws + (size_t)L_SEQ * H_DIM;          // 512*32
  float* t2    = ws + (size_t)2 * L_SEQ * H_DIM;      // 512*128*32 (8 MB)

  k_ln_proj<<<L_SEQ, 256, 0, stream>>>(state, ln_w, ln_b, w_left, b_left,
                                       w_right, b_right, left, right);
  k_t<<<dim3(16, L_SEQ), 256, 0, stream>>>(left, w_gate, t2);
  k_main<<<dim3(L_SEQ / 16, L_SEQ / 16), 256, 0, stream>>>(
      xyz, t2, right, w_rbf, b_rbf, b_gate, out);
}